// ExpertMLPLoRA_13056700580556
// MI455X (gfx1250) — compile-verified
//
#include <hip/hip_runtime.h>

// ---------------------------------------------------------------------------
// ExpertMLP LoRA: out[b,k,m,d] = SCALE * (z[b,k] @ A[e]) @ B[e]
// e = expert_indices[k]; B=8, K=8, M=512, D=1024, R=16, SCALE=1/16.
//
// Memory-bound (AI ~= 8 flop/byte; 268 MB one-shot traffic -> ~11.5us floor
// at 23.3 TB/s). Full-precision f32 via v_wmma_f32_16x16x4_f32. LoRA factors
// (128KB/expert) live in LDS; z streamed once (NT loads, register-prefetch
// software pipeline), out written once (NT stores).
// ---------------------------------------------------------------------------

typedef float v2f __attribute__((ext_vector_type(2)));
typedef float v4f __attribute__((ext_vector_type(4)));
typedef float v8f __attribute__((ext_vector_type(8)));

#define DIMD     1024
#define RANK     16
#define MROWS    512
#define NBATCH   8
#define NK       8
#define LSCALE   (1.0f / 16.0f)

#define MTILES   4      // M tiles per (b,k) slice -> 128 rows per block
#define KCHUNK   64     // z streamed in 16x64 chunks per wave
#define ZSTRIDE  68     // padded z row stride (floats): conflict-free, 16B-aligned
#define BSTRIDE  1040   // padded B_lora row stride: +32 bank offset between halves
#define DSTRIDE  18     // padded down-tile row stride: 8B-aligned, conflict-free

__global__ __launch_bounds__(256)
void lora_expert_wmma_f32(const float* __restrict__ z,
                          const float* __restrict__ A_all,
                          const float* __restrict__ B_all,
                          const int*   __restrict__ eidx,
                          float*       __restrict__ out)
{
    __shared__ float lds_A[DIMD * RANK];          // A_e[d][r], r fastest   (64 KB)
    __shared__ float lds_B[RANK * BSTRIDE];       // B_e[r][d], padded      (65 KB)
    __shared__ float lds_z[8 * 16 * ZSTRIDE];     // per-wave z staging     (34 KB)
    __shared__ float lds_down[8 * 16 * DSTRIDE];  // per-wave down relayout  (9 KB)

    const int tid  = threadIdx.x;
    const int wave = tid >> 5;
    const int lane = tid & 31;
    const int half = lane >> 4;   // 0: lanes 0-15, 1: lanes 16-31
    const int l    = lane & 15;

    const int bid = blockIdx.x;
    const int mt  = bid & (MTILES - 1);
    const int bk  = bid >> 2;                 // (b*NK + k), MTILES == 4
    const int k   = bk & (NK - 1);
    const int e   = eidx[k];

    const float* Ae = A_all + (size_t)e * DIMD * RANK;
    const float* Be = B_all + (size_t)e * RANK * DIMD;
    const float* zs = z   + (size_t)bk * MROWS * DIMD;
    float*       os = out + (size_t)bk * MROWS * DIMD;

    // ---- cooperative load of A_e (natural) and B_e (row-padded) into LDS ----
    #pragma unroll
    for (int i = 0; i < 16; ++i) {
        int idx = (i * 256 + tid) * 4;                       // v4f granules
        *(v4f*)&lds_A[idx] = *(const v4f*)&Ae[idx];
        int r = idx >> 10, c = idx & 1023;
        *(v4f*)&lds_B[r * BSTRIDE + c] = *(const v4f*)&Be[idx];
    }
    __syncthreads();

    const int mrow0 = mt * 128 + wave * 16;   // base row of this wave's strip

    float* zbuf = &lds_z[wave * 16 * ZSTRIDE];
    float* dbuf = &lds_down[wave * 16 * DSTRIDE];

    // per-lane staging geometry: i-th v4f covers (row, col) of the 16x64 chunk
    const int prow = (lane * 4) >> 6;         // 0 or 1 within each i-pair
    const int pcol = (lane * 4) & 63;

    // ---- phase 1: down[16,16] = z_strip[16,1024] @ A_e[1024,16] ----
    // Software pipeline: pf[] holds the next chunk while WMMAs chew the
    // current one (DS ops are in-order per wave -> single LDS buffer is safe).
    v8f acc0 = {0.f,0.f,0.f,0.f,0.f,0.f,0.f,0.f};
    v8f acc1 = {0.f,0.f,0.f,0.f,0.f,0.f,0.f,0.f};

    v4f pf[8];
    #pragma unroll
    for (int i = 0; i < 8; ++i) {
        int row = i * 2 + prow;
        pf[i] = __builtin_nontemporal_load(
                    (const v4f*)&zs[(size_t)(mrow0 + row) * DIMD + pcol]);
    }

    for (int kc = 0; kc < DIMD; kc += KCHUNK) {
        // commit staged chunk to LDS
        #pragma unroll
        for (int i = 0; i < 8; ++i) {
            int row = i * 2 + prow;
            *(v4f*)&zbuf[row * ZSTRIDE + pcol] = pf[i];
        }
        // issue next chunk's loads early (overlap with WMMAs below)
        if (kc + KCHUNK < DIMD) {
            #pragma unroll
            for (int i = 0; i < 8; ++i) {
                int row = i * 2 + prow;
                pf[i] = __builtin_nontemporal_load(
                            (const v4f*)&zs[(size_t)(mrow0 + row) * DIMD
                                            + (kc + KCHUNK) + pcol]);
            }
        }
        // 16 WMMAs over this chunk, two independent accumulator chains
        #pragma unroll
        for (int kk = 0; kk < KCHUNK; kk += 8) {
            // A-frag (z): lanes 0-15 -> K = kk,kk+1 ; lanes 16-31 -> K = kk+2,kk+3
            v2f a0 = *(const v2f*)&zbuf[l * ZSTRIDE + kk + 2 * half];
            int d0 = kc + kk + 2 * half;
            v2f b0;
            b0.x = lds_A[d0 * RANK + l];
            b0.y = lds_A[(d0 + 1) * RANK + l];
            acc0 = __builtin_amdgcn_wmma_f32_16x16x4_f32(
                       false, a0, false, b0, (short)0, acc0, false, false);

            v2f a1 = *(const v2f*)&zbuf[l * ZSTRIDE + kk + 4 + 2 * half];
            int d1 = d0 + 4;
            v2f b1;
            b1.x = lds_A[d1 * RANK + l];
            b1.y = lds_A[(d1 + 1) * RANK + l];
            acc1 = __builtin_amdgcn_wmma_f32_16x16x4_f32(
                       false, a1, false, b1, (short)0, acc1, false, false);
        }
    }

    // ---- relayout: C-layout (m in VGPR idx, r in lane) -> A-frag layout ----
    #pragma unroll
    for (int j = 0; j < 8; ++j) {
        int m = j + 8 * half;                 // C/D layout: VGPR j -> M = j / j+8
        dbuf[m * DSTRIDE + l] = (acc0[j] + acc1[j]) * LSCALE;
    }

    // ---- phase 2: up[16,1024] = down[16,16] @ B_e[16,1024], streamed out ----
    #pragma unroll 2
    for (int nt = 0; nt < DIMD / 16; ++nt) {
        v8f c2 = {0.f,0.f,0.f,0.f,0.f,0.f,0.f,0.f};
        #pragma unroll
        for (int kk = 0; kk < RANK; kk += 4) {
            v2f a2 = *(const v2f*)&dbuf[l * DSTRIDE + kk + 2 * half];
            int r = kk + 2 * half;
            v2f b2;
            b2.x = lds_B[r * BSTRIDE + nt * 16 + l];
            b2.y = lds_B[(r + 1) * BSTRIDE + nt * 16 + l];
            c2 = __builtin_amdgcn_wmma_f32_16x16x4_f32(
                     false, a2, false, b2, (short)0, c2, false, false);
        }
        #pragma unroll
        for (int j = 0; j < 8; ++j) {
            int m = mrow0 + j + 8 * half;
            __builtin_nontemporal_store(c2[j], &os[(size_t)m * DIMD + nt * 16 + l]);
        }
    }
}

extern "C" void kernel_launch(void* const* d_in, const int* in_sizes, int n_in,
                              void* d_out, int out_size, void* d_ws, size_t ws_size,
                              hipStream_t stream) {
    (void)in_sizes; (void)n_in; (void)out_size; (void)d_ws; (void)ws_size;
    const float* z     = (const float*)d_in[0];
    const float* A_all = (const float*)d_in[1];
    const float* B_all = (const float*)d_in[2];
    const int*   eidx  = (const int*)d_in[3];
    float*       out   = (float*)d_out;

    dim3 grid(NBATCH * NK * MTILES);   // 256 workgroups
    dim3 block(256);                   // 8 wave32 per workgroup
    lora_expert_wmma_f32<<<grid, block, 0, stream>>>(z, A_all, B_all, eidx, out);
}